// GCN_1262720385649
// MI455X (gfx1250) — compile-verified
//
#include <hip/hip_runtime.h>
#include <cstddef>

// GCN 3-layer forward for MI455X (gfx1250, wave32).
//   layer: agg = D^-1/2 (A+I) D^-1/2 (X @ W) + b ; relu between layers.
// GEMMs use V_WMMA_F32_16X16X4_F32 (full f32 precision, matches reference).
// Scatter-add uses hardware global_atomic_add_f32 (unsafeAtomicAdd).
// ReLU is fused into the next layer's GEMM A-tile load (no extra pass).

#define N_NODES 100000
#define N_EDGES 1600000
#define IN_DIM  128
#define HID_DIM 64

typedef float v2f __attribute__((ext_vector_type(2)));
typedef float v8f __attribute__((ext_vector_type(8)));

__global__ void zero_f32(float* __restrict__ p, int n) {
    int i = blockIdx.x * blockDim.x + threadIdx.x;
    if (i < n) p[i] = 0.0f;
}

// deg[dst] += 1 over all edges (float, exact for these magnitudes)
__global__ void deg_count(const int* __restrict__ dst, float* __restrict__ deg, int e) {
    int i = blockIdx.x * blockDim.x + threadIdx.x;
    if (i < e) unsafeAtomicAdd(&deg[dst[i]], 1.0f);
}

// dis[i] = rsqrt(deg[i] + 1)   (in place; +1 = self loop)
__global__ void finalize_dis(float* __restrict__ dis, int n) {
    int i = blockIdx.x * blockDim.x + threadIdx.x;
    if (i < n) dis[i] = rsqrtf(dis[i] + 1.0f);
}

// H[tile16 x 64] = (relu?)(X[tile16 x K]) @ W[K x 64]
// 4 waves/block, wave w owns output columns [16w, 16w+16).
// A-fragment layout (ISA 7.12.2, 32-bit A 16x4): lanes 0-15 hold K=k0,k0+1;
// lanes 16-31 hold K=k0+2,k0+3 for row M = lane&15.
// C/D layout: VGPR r -> M=r (lanes 0-15) / M=r+8 (lanes 16-31), N = lane&15.
template <int K, bool RELU>
__global__ __launch_bounds__(128) void gemm_wmma(const float* __restrict__ X,
                                                 const float* __restrict__ W,
                                                 float* __restrict__ H) {
    __shared__ float ldsA[16 * (K + 1)];   // +1 pad: stride K+1 ≡ 1 (mod 64) -> no bank conflicts
    const int tid  = threadIdx.x;
    const int tile = blockIdx.x;

    // Stage A tile (16 x K, contiguous rows) into LDS, fusing ReLU of prev layer.
    const float* Ablk = X + (size_t)tile * 16 * K;
    for (int idx = tid; idx < 16 * K; idx += 128) {
        float v = Ablk[idx];
        if (RELU) v = fmaxf(v, 0.0f);
        ldsA[(idx / K) * (K + 1) + (idx % K)] = v;
    }
    __syncthreads();

    const int lane = tid & 31;
    const int wv   = tid >> 5;
    const int m    = lane & 15;
    const int koff = (lane >> 4) << 1;          // 0 for lanes 0-15, 2 for lanes 16-31
    const int n    = (wv << 4) + (lane & 15);   // output column

    v8f acc = {};
    for (int k0 = 0; k0 < K; k0 += 4) {
        v2f a, b;
        a.x = ldsA[m * (K + 1) + k0 + koff];
        a.y = ldsA[m * (K + 1) + k0 + koff + 1];
        b.x = W[(k0 + koff)     * HID_DIM + n];  // B 4x16: row K striped like A halves
        b.y = W[(k0 + koff + 1) * HID_DIM + n];
        acc = __builtin_amdgcn_wmma_f32_16x16x4_f32(
            /*neg_a=*/false, a, /*neg_b=*/false, b,
            /*c_mod=*/(short)0, acc, /*reuse_a=*/false, /*reuse_b=*/false);
    }

    float* Hblk = H + (size_t)tile * 16 * HID_DIM;
    const int rbase = (lane >> 4) << 3;          // M offset 0 or 8
#pragma unroll
    for (int r = 0; r < 8; ++r)
        Hblk[(rbase + r) * HID_DIM + n] = acc[r];
}

// out[i,f] = H[i,f] * dis[i]^2 + b[f]    (self-loop term + bias)
__global__ void agg_init(const float* __restrict__ H, const float* __restrict__ dis,
                         const float* __restrict__ b, float* __restrict__ out, int n) {
    int gid = blockIdx.x * blockDim.x + threadIdx.x;
    if (gid >= n * HID_DIM) return;
    int node = gid >> 6;
    int f    = gid & 63;
    float d  = dis[node];
    out[gid] = fmaf(H[gid], d * d, b[f]);
}

// out[dst,f] += H[src,f] * dis[src]*dis[dst]   — one thread per (edge,feature);
// 64 consecutive threads share an edge -> coalesced 128B gathers per wave and
// contiguous atomic bursts into one/two L2 lines.
__global__ void agg_edges(const float* __restrict__ H, const float* __restrict__ dis,
                          const int* __restrict__ src, const int* __restrict__ dst,
                          float* __restrict__ out) {
    int gid = blockIdx.x * blockDim.x + threadIdx.x;   // < 1.6M*64 = 102.4M, fits int
    int e = gid >> 6;
    if (e >= N_EDGES) return;
    int f = gid & 63;
    int s = src[e];
    int d = dst[e];
    float norm = dis[s] * dis[d];
    unsafeAtomicAdd(&out[d * HID_DIM + f], H[s * HID_DIM + f] * norm);
}

extern "C" void kernel_launch(void* const* d_in, const int* in_sizes, int n_in,
                              void* d_out, int out_size, void* d_ws, size_t ws_size,
                              hipStream_t stream) {
    const float* x  = (const float*)d_in[0];
    const int*   ei = (const int*)  d_in[1];
    const float* W1 = (const float*)d_in[2];
    const float* b1 = (const float*)d_in[3];
    const float* W2 = (const float*)d_in[4];
    const float* b2 = (const float*)d_in[5];
    const float* W3 = (const float*)d_in[6];
    const float* b3 = (const float*)d_in[7];
    const int* src = ei;            // edge_index[0]
    const int* dst = ei + N_EDGES;  // edge_index[1]

    float* ws  = (float*)d_ws;
    float* dis = ws;                                   // N floats (deg -> rsqrt)
    float* H   = ws + N_NODES;                         // N x 64 (GEMM out)
    float* AGG = H + (size_t)N_NODES * HID_DIM;        // N x 64 (aggregated)
    float* out = (float*)d_out;

    const int T = 256;
    const int gN  = (N_NODES + T - 1) / T;
    const int gE  = (N_EDGES + T - 1) / T;
    const int gNF = (N_NODES * HID_DIM) / T;           // 25000 exact
    const int gEF = (N_EDGES * HID_DIM) / T;           // 400000 exact
    const int gT  = N_NODES / 16;                      // 6250 tiles exact

    // degrees -> deg_inv_sqrt
    zero_f32<<<gN, T, 0, stream>>>(dis, N_NODES);
    deg_count<<<gE, T, 0, stream>>>(dst, dis, N_EDGES);
    finalize_dis<<<gN, T, 0, stream>>>(dis, N_NODES);

    // layer 1: H = x @ W1 ; AGG = scatter + selfloop + b1   (relu deferred into next GEMM)
    gemm_wmma<IN_DIM, false><<<gT, 128, 0, stream>>>(x, W1, H);
    agg_init<<<gNF, T, 0, stream>>>(H, dis, b1, AGG, N_NODES);
    agg_edges<<<gEF, T, 0, stream>>>(H, dis, src, dst, AGG);

    // layer 2: H = relu(AGG) @ W2 ; AGG = ...
    gemm_wmma<HID_DIM, true><<<gT, 128, 0, stream>>>(AGG, W2, H);
    agg_init<<<gNF, T, 0, stream>>>(H, dis, b2, AGG, N_NODES);
    agg_edges<<<gEF, T, 0, stream>>>(H, dis, src, dst, AGG);

    // layer 3: out = relu(AGG) @ W3 aggregated, no relu at the end
    gemm_wmma<HID_DIM, true><<<gT, 128, 0, stream>>>(AGG, W3, H);
    agg_init<<<gNF, T, 0, stream>>>(H, dis, b3, out, N_NODES);
    agg_edges<<<gEF, T, 0, stream>>>(H, dis, src, dst, out);
}